// Attention_18631568130118
// MI455X (gfx1250) — compile-verified
//
#include <hip/hip_runtime.h>

#define SEQ_L   4096
#define DIM     128
#define THREADS 256
#define WAVES   8
#define GROUPS  (SEQ_L / 16)   // 256 groups of 16 keys

typedef __attribute__((ext_vector_type(2))) float v2f;
typedef __attribute__((ext_vector_type(8))) float v8f;

__global__ __launch_bounds__(THREADS)
void Attention_18631568130118_kernel(const float* __restrict__ q,
                                     const float* __restrict__ k,
                                     const int*   __restrict__ mask,
                                     float*       __restrict__ out)
{
    __shared__ float s_scores[SEQ_L];   // 16 KB score buffer
    __shared__ float s_red[WAVES];      // per-wave reduction scratch

    const int bh   = blockIdx.x;        // 0..255 = b*16+h
    const int b    = bh >> 4;
    const int tid  = threadIdx.x;
    const int lane = tid & 31;
    const int wave = tid >> 5;          // 0..7
    const int lrow = lane & 15;         // key-in-group / N column
    const int hsel = lane >> 4;         // lane-half selects K pair {0,1} vs {2,3}

    const float* qv   = q    + (size_t)bh * DIM;
    const float* kb   = k    + (size_t)bh * SEQ_L * DIM;
    const int*   mrow = mask + (size_t)b  * SEQ_L;

    // ---- query norm (redundant per wave, cheap) ----
    float qs = 0.f;
    #pragma unroll
    for (int d = 0; d < DIM; d += 32) {
        float v = qv[d + lane];
        qs += v * v;
    }
    #pragma unroll
    for (int off = 16; off >= 1; off >>= 1) qs += __shfl_xor(qs, off, 32);
    const float qn = sqrtf(qs);

    // ---- preload A operand: q replicated across all 16 M rows ----
    // A 16x4 f32 layout: VGPR0 = K(2*hsel), VGPR1 = K(2*hsel+1), same for all M.
    v2f a_reg[32];
    #pragma unroll
    for (int c = 0; c < 32; ++c) {
        a_reg[c].x = qv[4 * c + 2 * hsel];
        a_reg[c].y = qv[4 * c + 2 * hsel + 1];
    }

    // ---- phase 1: scores via WMMA, streamed key reads ----
    float lmax = -3.0e38f;
    for (int g = wave; g < GROUPS; g += WAVES) {
        // lane L streams key row (g*16 + L&15), dims 2*hsel..(+1) mod 4
        const float* kp = kb + (size_t)(g * 16 + lrow) * DIM + 2 * hsel;
        if (g + WAVES < GROUPS)
            __builtin_prefetch(kp + (size_t)WAVES * 16 * DIM, 0, 0);

        v8f   acc = {};
        float nsq = 0.f;
        #pragma unroll
        for (int c = 0; c < 32; ++c) {
            v2f bvec;
            bvec.x = kp[4 * c];       // K = 2*hsel   of chunk c
            bvec.y = kp[4 * c + 1];   // K = 2*hsel+1 of chunk c
            nsq += bvec.x * bvec.x + bvec.y * bvec.y;
            acc = __builtin_amdgcn_wmma_f32_16x16x4_f32(
                      false, a_reg[c], false, bvec, (short)0, acc, false, false);
        }
        // combine lane-halves: full ||k||^2 for key (lane&15)
        nsq += __shfl_xor(nsq, 16, 32);

        const float dot   = acc[0];                 // D[M][lane&15], all M rows equal
        const float kn    = sqrtf(nsq);
        float score       = dot / fmaxf(qn * kn, 1e-8f);
        const int  keyidx = g * 16 + lrow;
        if (mrow[keyidx] == 0) score = -1.0e9f;
        lmax = fmaxf(lmax, score);
        if (hsel == 0) s_scores[keyidx] = score;
    }

    // ---- block max ----
    #pragma unroll
    for (int off = 16; off >= 1; off >>= 1) lmax = fmaxf(lmax, __shfl_xor(lmax, off, 32));
    if (lane == 0) s_red[wave] = lmax;
    __syncthreads();
    float gmax = s_red[0];
    #pragma unroll
    for (int w = 1; w < WAVES; ++w) gmax = fmaxf(gmax, s_red[w]);
    __syncthreads();  // protect s_red reuse

    // ---- phase 2: exp + block sum ----
    float lsum = 0.f;
    for (int l = tid; l < SEQ_L; l += THREADS) {
        float e = __expf(s_scores[l] - gmax);
        s_scores[l] = e;
        lsum += e;
    }
    #pragma unroll
    for (int off = 16; off >= 1; off >>= 1) lsum += __shfl_xor(lsum, off, 32);
    if (lane == 0) s_red[wave] = lsum;
    __syncthreads();
    float gsum = 0.f;
    #pragma unroll
    for (int w = 0; w < WAVES; ++w) gsum += s_red[w];
    const float inv = 1.0f / gsum;

    // ---- phase 3: normalize, coalesced store ----
    float* orow = out + (size_t)bh * SEQ_L;
    for (int l = tid; l < SEQ_L; l += THREADS)
        orow[l] = s_scores[l] * inv;
}

extern "C" void kernel_launch(void* const* d_in, const int* in_sizes, int n_in,
                              void* d_out, int out_size, void* d_ws, size_t ws_size,
                              hipStream_t stream) {
    (void)in_sizes; (void)n_in; (void)out_size; (void)d_ws; (void)ws_size;
    const float* q    = (const float*)d_in[0];
    const float* kk   = (const float*)d_in[1];
    const int*   mask = (const int*)d_in[2];
    float*       out  = (float*)d_out;

    dim3 grid(256);   // B*H = 16*16
    dim3 block(THREADS);
    Attention_18631568130118_kernel<<<grid, block, 0, stream>>>(q, kk, mask, out);
}